// IVPLoss_65403761983664
// MI455X (gfx1250) — compile-verified
//
#include <hip/hip_runtime.h>

// Problem constants (match reference)
#define BB       8
#define HH       512
#define WW       512
#define NSTEPS   8
#define DXC      0.5f
#define NPIX     (BB * HH * WW)          // 2,097,152
#define THREADS  256
#define PIXPT    4                        // pixels per thread (MLP)
#define NBLOCKS  (NPIX / (THREADS * PIXPT))  // 2048
#define STRIDE   (NBLOCKS * THREADS)         // 524,288
#define TOTAL_ELEMS 18874368.0f           // 9 * 8 * 512 * 512

typedef float v2f __attribute__((ext_vector_type(2)));
typedef float v8f __attribute__((ext_vector_type(8)));

// Bilinear interp of one (2,H,W) planar field at clipped (x,y).
__device__ __forceinline__ v2f bilerp(const float* __restrict__ f, float x, float y) {
    x = fminf(fmaxf(x, 0.0f), (float)(WW - 1));
    y = fminf(fmaxf(y, 0.0f), (float)(HH - 1));
    float x0f = floorf(x), y0f = floorf(y);
    float wx = x - x0f, wy = y - y0f;
    int x0 = (int)x0f, y0 = (int)y0f;
    int x1 = min(x0 + 1, WW - 1);
    int y1 = min(y0 + 1, HH - 1);
    int i00 = y0 * WW + x0;
    int i01 = y0 * WW + x1;
    int i10 = y1 * WW + x0;
    int i11 = y1 * WW + x1;
    const float* fx = f;            // channel 0 (x velocity)
    const float* fy = f + HH * WW;  // channel 1 (y velocity)
    float a00 = fx[i00], a01 = fx[i01], a10 = fx[i10], a11 = fx[i11];
    float b00 = fy[i00], b01 = fy[i01], b10 = fy[i10], b11 = fy[i11];
    float tx0 = a00 + wx * (a01 - a00);
    float tx1 = a10 + wx * (a11 - a10);
    float ty0 = b00 + wx * (b01 - b00);
    float ty1 = b10 + wx * (b11 - b10);
    v2f r;
    r.x = tx0 + wy * (tx1 - tx0);
    r.y = ty0 + wy * (ty1 - ty0);
    return r;
}

// Sum v over all 32 lanes of the wave using V_WMMA_F32_16X16X4_F32:
// A[M=lane%16, K=(lane<16?0:2)] = v, other K = 0; B = ones -> D rows = row sums.
// Every lane returns the full 32-lane sum.
__device__ __forceinline__ float wave_sum32_wmma(float v) {
    v2f a; a.x = v;    a.y = 0.0f;
    v2f b; b.x = 1.0f; b.y = 1.0f;
    v8f c = {};
    v8f d = __builtin_amdgcn_wmma_f32_16x16x4_f32(
        /*neg_a=*/false, a, /*neg_b=*/false, b,
        /*c_mod=*/(short)0, c, /*reuse_a=*/false, /*reuse_b=*/false);
    float s = d[0] + d[1] + d[2] + d[3] + d[4] + d[5] + d[6] + d[7];
    s += __shfl_xor(s, 16, 32);   // fold lanes 0-15 (M=0..7) with 16-31 (M=8..15)
    return s;
}

__global__ void __launch_bounds__(THREADS)
ivp_main(const float* __restrict__ vf_pred, const float* __restrict__ vf_true,
         float* __restrict__ partial) {
    const int tid = blockIdx.x * THREADS + threadIdx.x;

    // 4 independent trajectory pairs per thread -> 64 independent gathers/step.
    float pxt[PIXPT], pyt[PIXPT], pxp[PIXPT], pyp[PIXPT];
    const float* ft[PIXPT];
    const float* fp[PIXPT];

#pragma unroll
    for (int k = 0; k < PIXPT; ++k) {
        int g = tid + k * STRIDE;
        int b = g >> 18;            // g / (H*W)
        int y = (g >> 9) & (HH - 1);
        int x = g & (WW - 1);
        ft[k] = vf_true + (size_t)b * 2u * HH * WW;
        fp[k] = vf_pred + (size_t)b * 2u * HH * WW;
        pxt[k] = (float)x; pyt[k] = (float)y;
        pxp[k] = (float)x; pyp[k] = (float)y;
    }

    float err = 0.0f;
    for (int s = 0; s < NSTEPS; ++s) {
        v2f vt[PIXPT], vp[PIXPT];
#pragma unroll
        for (int k = 0; k < PIXPT; ++k) {   // issue all gathers first (MLP)
            vt[k] = bilerp(ft[k], pxt[k], pyt[k]);
            vp[k] = bilerp(fp[k], pxp[k], pyp[k]);
        }
#pragma unroll
        for (int k = 0; k < PIXPT; ++k) {
            pxt[k] = fmaf(DXC, vt[k].x, pxt[k]);
            pyt[k] = fmaf(DXC, vt[k].y, pyt[k]);
            pxp[k] = fmaf(DXC, vp[k].x, pxp[k]);
            pyp[k] = fmaf(DXC, vp[k].y, pyp[k]);
            float dx = pxt[k] - pxp[k];
            float dy = pyt[k] - pyp[k];
            err = fmaf(dx, dx, err);
            err = fmaf(dy, dy, err);
        }
    }

    // Wave-level reduction via WMMA (EXEC all ones: no divergence up to here).
    float wsum = wave_sum32_wmma(err);

    __shared__ float lds[THREADS / 32];
    const int lane = threadIdx.x & 31;
    const int wid  = threadIdx.x >> 5;
    if (lane == 0) lds[wid] = wsum;
    __syncthreads();
    if (threadIdx.x == 0) {
        float t = 0.0f;
#pragma unroll
        for (int i = 0; i < THREADS / 32; ++i) t += lds[i];
        partial[blockIdx.x] = t;
    }
}

// Single-wave final reduction of NBLOCKS partials with chained D->C WMMA
// accumulation (64 values per wmma). Deterministic order.
__global__ void __launch_bounds__(32)
ivp_reduce(const float* __restrict__ partial, float* __restrict__ out) {
    const int lane = threadIdx.x;   // 0..31, EXEC all ones
    v2f bones; bones.x = 1.0f; bones.y = 1.0f;
    v8f c = {};
    for (int i = 0; i < NBLOCKS; i += 64) {
        v2f a;
        a.x = partial[i + lane];
        a.y = partial[i + 32 + lane];
        c = __builtin_amdgcn_wmma_f32_16x16x4_f32(
            false, a, false, bones, (short)0, c, false, false);
    }
    float s = c[0] + c[1] + c[2] + c[3] + c[4] + c[5] + c[6] + c[7];
    s += __shfl_xor(s, 16, 32);
    if (lane == 0) out[0] = s * (1.0f / TOTAL_ELEMS);
}

extern "C" void kernel_launch(void* const* d_in, const int* in_sizes, int n_in,
                              void* d_out, int out_size, void* d_ws, size_t ws_size,
                              hipStream_t stream) {
    const float* vf_pred = (const float*)d_in[0];
    const float* vf_true = (const float*)d_in[1];
    float* partial = (float*)d_ws;          // NBLOCKS floats = 8 KB
    float* out = (float*)d_out;

    ivp_main<<<NBLOCKS, THREADS, 0, stream>>>(vf_pred, vf_true, partial);
    ivp_reduce<<<1, 32, 0, stream>>>(partial, out);
}